// TargetExtractionBlock_13632226198116
// MI455X (gfx1250) — compile-verified
//
#include <hip/hip_runtime.h>

// ---------------------------------------------------------------------------
// TargetExtractionBlock on MI455X (gfx1250):
//   GEMM1 (32768x768 @ 768x256, fp32) via V_WMMA_F32_16X16X4_F32,
//   X-tile staged with GLOBAL_LOAD_ASYNC_TO_LDS_B128 (ASYNCcnt DMA),
//   + bias/ReLU + tiny GEMM2 (x W2[256x5]) fused, then per-batch
//   boundary add + Viterbi decode.
// fp32 WMMA chosen deliberately: potentials is a directly-compared fp32
// output feeding an argmax chain; bf16 would be ~8x matrix rate (HBM floor
// ~4.3us for the 101MB activation read) but perturbs 768-deep reductions.
// ---------------------------------------------------------------------------

typedef __attribute__((ext_vector_type(2))) float v2f;
typedef __attribute__((ext_vector_type(8))) float v8f;

#define BA   128      // B*A flattened batch
#define SEQ  256
#define HDIM 768
#define HID  256
#define NC   5
#define LDS_STRIDE 772   // 768 + 4 pad: row-reads land on distinct banks
#define HSTRIDE    260   // 256 + 4 pad for h-tile

// d_out layout (floats)
#define OUT_DEC   0
#define OUT_POT   (BA * SEQ)
#define OUT_SEQ   (OUT_POT + BA * SEQ * NC)
#define OUT_CHAIN (OUT_SEQ + BA)

// ---------------------------------------------------------------------------
// Kernel 1: one block per 16-row M-tile. 16 waves; wave w computes the
// 16x16 output tile at columns [16w, 16w+16) of h = relu(X@W1 + b1),
// then 80 threads fold the h-tile through W2 (+b2) into potentials.
// ---------------------------------------------------------------------------
__global__ __launch_bounds__(512)
void gemm_fused_kernel(const float* __restrict__ X,    // [BA*SEQ, HDIM]
                       const float* __restrict__ W1,   // [HDIM, HID]
                       const float* __restrict__ b1,   // [HID]
                       const float* __restrict__ W2,   // [HID, NC]
                       const float* __restrict__ b2,   // [NC]
                       float* __restrict__ pot)        // [BA*SEQ, NC]
{
    __shared__ float lds[16 * LDS_STRIDE];             // 49408 B; reused for h

    const int tile = blockIdx.x;                       // 0..2047
    const int tid  = threadIdx.x;
    const int wave = tid >> 5;
    const int lane = tid & 31;

    // ---- stage X tile (16 x 768 f32 = 48KB) into LDS via async DMA ----
    // 512 threads x 6 chunks x 16B. A B128 chunk (4 floats) never straddles
    // a 768-float row, so the padded LDS layout stays contiguous per chunk.
    {
        const float* xbase = X + (size_t)tile * 16 * HDIM;
        #pragma unroll
        for (int i = 0; i < 6; ++i) {
            int e = (tid + i * 512) * 4;               // element index, mult of 4
            unsigned goff = (unsigned)(e * 4);         // global byte offset
            unsigned loff = (unsigned)(size_t)
                &lds[(e / HDIM) * LDS_STRIDE + (e % HDIM)];   // LDS byte addr
            asm volatile("global_load_async_to_lds_b128 %0, %1, %2"
                         :
                         : "v"(loff), "v"(goff), "s"(xbase)
                         : "memory");
        }
        asm volatile("s_wait_asynccnt 0" ::: "memory");
    }
    __syncthreads();

    // ---- WMMA fragment addressing (ISA 7.12.2, 32-bit 16x4 A-layout):
    //   lanes 0-15 : M=lane,    v0=K+0, v1=K+1
    //   lanes 16-31: M=lane-16, v0=K+2, v1=K+3   (mirrored for B 4x16)
    const int n0    = wave * 16;
    const int mrow  = lane & 15;
    const int khalf = lane >> 4;                       // 0 or 1
    const int n     = n0 + (lane & 15);

    const float* xl = &lds[mrow * LDS_STRIDE + 2 * khalf];
    const float* wl = W1 + (size_t)(2 * khalf) * HID + n;

    v8f acc = {};
    #pragma unroll 8
    for (int k = 0; k < HDIM; k += 4) {
        v2f a, b;
        a.x = xl[k];
        a.y = xl[k + 1];
        b.x = wl[(size_t)k * HID];                     // W1[k+2*khalf][n]
        b.y = wl[(size_t)(k + 1) * HID];               // W1[k+2*khalf+1][n]
        __builtin_prefetch(&wl[(size_t)(k + 16) * HID], 0, 3); // WGP-scope
        acc = __builtin_amdgcn_wmma_f32_16x16x4_f32(
                  /*neg_a=*/false, a, /*neg_b=*/false, b,
                  /*c_mod=*/(short)0, acc, /*reuse_a=*/false, /*reuse_b=*/false);
    }

    // ---- bias + ReLU, park h-tile in LDS (reuse buffer after barrier) ----
    const float bias = b1[n];
    __syncthreads();                                   // all X reads done
    #pragma unroll
    for (int j = 0; j < 8; ++j) {
        // D layout: VGPR j holds row M = j + 8*(lane>>4), column n
        int   m = j + 8 * khalf;
        float v = acc[j] + bias;
        lds[m * HSTRIDE + n] = v > 0.0f ? v : 0.0f;
    }
    __syncthreads();

    // ---- GEMM2: 16 rows x 5 classes, one thread each ----
    if (tid < 16 * NC) {
        int m = tid / NC, c = tid % NC;
        float s = b2[c];
        #pragma unroll 8
        for (int i = 0; i < HID; ++i)
            s += lds[m * HSTRIDE + i] * W2[i * NC + c];
        pot[((size_t)tile * 16 + m) * NC + c] = s;
    }
}

// ---------------------------------------------------------------------------
// Kernel 2: one thread per batch. seq_len from mask, boundary adds,
// Viterbi forward scan (backpointers in d_ws), backtrack, passthrough outs.
// ---------------------------------------------------------------------------
__global__ __launch_bounds__(128)
void viterbi_kernel(const unsigned char* __restrict__ mask, // [BA, SEQ] bool
                    const float* __restrict__ chain,        // [NC, NC]
                    const float* __restrict__ lb,           // [NC]
                    const float* __restrict__ rb,           // [NC]
                    float* __restrict__ out,                // full d_out
                    signed char* __restrict__ bp_ws)        // [BA, SEQ-1, NC]
{
    const int b = threadIdx.x;
    if (b >= BA) return;

    float* dec   = out + OUT_DEC;
    float* pot   = out + OUT_POT;
    float* seqo  = out + OUT_SEQ;
    float* chout = out + OUT_CHAIN;

    if (b < NC * NC) chout[b] = chain[b];

    // seq_len = sum(mask[b])
    int sl = 0;
    for (int s = 0; s < SEQ; ++s) sl += (mask[b * SEQ + s] != 0) ? 1 : 0;
    seqo[b] = (float)sl;

    float* pb   = pot + (size_t)b * SEQ * NC;
    int    last = min(max(sl - 1, 0), SEQ - 1);
    #pragma unroll
    for (int c = 0; c < NC; ++c) {
        pb[c]             += lb[c];
        pb[last * NC + c] += rb[c];
    }

    float ck[NC * NC];
    #pragma unroll
    for (int i = 0; i < NC * NC; ++i) ck[i] = chain[i];

    float score[NC];
    #pragma unroll
    for (int c = 0; c < NC; ++c) score[c] = pb[c];

    signed char* bp = bp_ws + (size_t)b * (SEQ - 1) * NC;
    for (int s = 1; s < SEQ; ++s) {
        const bool valid = s < sl;
        float ns[NC];
        #pragma unroll
        for (int j = 0; j < NC; ++j) {
            float best = score[0] + ck[j];
            int   bi   = 0;
            #pragma unroll
            for (int i = 1; i < NC; ++i) {
                float cand = score[i] + ck[i * NC + j];
                if (cand > best) { best = cand; bi = i; }   // first-max wins
            }
            bp[(s - 1) * NC + j] = (signed char)bi;
            ns[j] = best + pb[s * NC + j];
        }
        if (valid) {
            #pragma unroll
            for (int j = 0; j < NC; ++j) score[j] = ns[j];
        }
    }

    // argmax of final score (first max), then backtrack
    int tag = 0; float bestf = score[0];
    #pragma unroll
    for (int i = 1; i < NC; ++i)
        if (score[i] > bestf) { bestf = score[i]; tag = i; }

    dec[b * SEQ + (SEQ - 1)] = (float)tag;
    for (int t = SEQ - 2; t >= 0; --t) {
        const bool valid = (t + 1) < sl;
        int prev = valid ? (int)bp[t * NC + tag] : tag;
        dec[b * SEQ + t] = (float)prev;
        tag = prev;
    }
}

// ---------------------------------------------------------------------------
extern "C" void kernel_launch(void* const* d_in, const int* in_sizes, int n_in,
                              void* d_out, int out_size, void* d_ws, size_t ws_size,
                              hipStream_t stream) {
    const float*         hs   = (const float*)d_in[0];         // hidden_states
    const unsigned char* mask = (const unsigned char*)d_in[1]; // bool mask
    const float*         W1   = (const float*)d_in[2];
    const float*         b1   = (const float*)d_in[3];
    const float*         W2   = (const float*)d_in[4];
    const float*         b2   = (const float*)d_in[5];
    const float*         ck   = (const float*)d_in[6];
    const float*         lb   = (const float*)d_in[7];
    const float*         rb   = (const float*)d_in[8];

    float* out = (float*)d_out;

    gemm_fused_kernel<<<(BA * SEQ) / 16, 512, 0, stream>>>(
        hs, W1, b1, W2, b2, out + OUT_POT);

    viterbi_kernel<<<1, 128, 0, stream>>>(
        mask, ck, lb, rb, out, (signed char*)d_ws);
}